// BGCFLayer_38130719654352
// MI455X (gfx1250) — compile-verified
//
#include <hip/hip_runtime.h>
#include <math.h>

#define N_NODES 8192
#define DIM 64
#define BATCH_SZ 4096

typedef __attribute__((ext_vector_type(2))) float v2f;
typedef __attribute__((ext_vector_type(4))) float v4f;
typedef __attribute__((ext_vector_type(8))) float v8f;

__device__ __forceinline__ v8f wmma4(v2f a, v2f b, v8f c) {
  // D = A(16x4 f32) * B(4x16 f32) + C(16x16 f32)
  return __builtin_amdgcn_wmma_f32_16x16x4_f32(false, a, false, b, (short)0, c,
                                               false, false);
}

// ---------------------------------------------------------------------------
// 16-byte global -> LDS async staging (gfx1250 GLOBAL_LOAD_ASYNC_TO_LDS_B128,
// tracked by ASYNCcnt). VDST is the wave-relative LDS byte address = low 32
// bits of the generic pointer to a __shared__ object.
// ---------------------------------------------------------------------------
__device__ __forceinline__ void stage16(const float* __restrict__ g,
                                        float* __restrict__ l) {
  unsigned int loff = (unsigned int)(unsigned long long)l;
  unsigned long long ga = (unsigned long long)g;
  asm volatile("global_load_async_to_lds_b128 %0, %1, off"
               :
               : "v"(loff), "v"(ga)
               : "memory");
}

__device__ __forceinline__ void stage_wait() {
  asm volatile("s_wait_asynccnt 0x0" ::: "memory");
}

// ---------------------------------------------------------------------------
// Y[8192,64] = (X .* S?) @ W[64,64], optional tanh epilogue. 8 waves/block,
// one 16-row stripe per wave. W is staged once into LDS (16 KB) and shared by
// all 8 waves; B-fragments are ds reads.
// ---------------------------------------------------------------------------
template <int SCALED, int TANH>
__global__ __launch_bounds__(256) void gemm64_kernel(
    const float* __restrict__ X, const float* __restrict__ S,
    const float* __restrict__ W, float* __restrict__ Y) {
  __shared__ __align__(16) float wbuf[DIM * DIM];
  const int tid = threadIdx.x;
#pragma unroll
  for (int rr = 0; rr < 4; ++rr) {
    const int c = (tid + 256 * rr) * 4;
    stage16(W + c, wbuf + c);
  }
  stage_wait();
  __syncthreads();

  const int wave = tid >> 5;
  const int lane = tid & 31;
  const int m = lane & 15;  // A row / B-C column within tile
  const int h = lane >> 4;  // K half
  const int row0 = (blockIdx.x * 8 + wave) * 16;

  v8f acc[4] = {};
  const float* xrow = X + (size_t)(row0 + m) * DIM;
  const float* srow = S + (size_t)(row0 + m) * DIM;

#pragma unroll
  for (int ks = 0; ks < 16; ++ks) {
    const int k0 = 4 * ks + 2 * h;
    v2f a = *(const v2f*)(xrow + k0);
    if (SCALED) {
      v2f sv = *(const v2f*)(srow + k0);
      a.x *= sv.x;
      a.y *= sv.y;
    }
#pragma unroll
    for (int j = 0; j < 4; ++j) {
      v2f b;
      b.x = wbuf[(k0 + 0) * DIM + 16 * j + m];
      b.y = wbuf[(k0 + 1) * DIM + 16 * j + m];
      acc[j] = wmma4(a, b, acc[j]);
    }
  }

#pragma unroll
  for (int j = 0; j < 4; ++j) {
#pragma unroll
    for (int r = 0; r < 8; ++r) {
      float v = acc[j][r];
      if (TANH) v = tanhf(v);
      Y[(size_t)(row0 + r + 8 * h) * DIM + 16 * j + m] = v;
    }
  }
}

// ---------------------------------------------------------------------------
// out[r[e], :] += val[e] * x[c[e], :]  (edge scatter; L2-resident atomics)
// ---------------------------------------------------------------------------
__global__ __launch_bounds__(256) void spmm_kernel(
    const int* __restrict__ rows, const int* __restrict__ cols,
    const float* __restrict__ val, const float* __restrict__ x,
    float* __restrict__ out, int nedges) {
  int idx = blockIdx.x * 256 + threadIdx.x;
  if (idx >= nedges * 32) return;
  int e = idx >> 5;
  int p = (idx & 31) << 1;
  int r = rows[e];
  int c = cols[e];
  float v = val[e];
  v2f xv = *(const v2f*)(x + (size_t)c * DIM + p);
  unsafeAtomicAdd(out + (size_t)r * DIM + p + 0, v * xv.x);
  unsafeAtomicAdd(out + (size_t)r * DIM + p + 1, v * xv.y);
}

__global__ __launch_bounds__(256) void zero_kernel(float* __restrict__ p, int n) {
  int i = blockIdx.x * 256 + threadIdx.x;
  if (i < n) p[i] = 0.0f;
}

// ---------------------------------------------------------------------------
// O = softmax_row(Q @ K^T) @ V ; Q,K,V,O are [n,64] fp32. Flash-style online
// softmax. 4 waves/block share async-staged K/V tiles in LDS; each wave owns a
// 16-query stripe and a private 16x16 LDS transpose buffer for P.
// ---------------------------------------------------------------------------
__global__ __launch_bounds__(128) void attn_kernel(
    const float* __restrict__ Q, const float* __restrict__ K,
    const float* __restrict__ V, float* __restrict__ O, int n) {
  __shared__ __align__(16) float smem[3072];  // kbuf | vbuf | 4x pbuf
  float* kbuf = smem;
  float* vbuf = smem + 1024;
  const int tid = threadIdx.x;
  const int wave = tid >> 5;
  const int lane = tid & 31;
  const int m = lane & 15;
  const int h = lane >> 4;
  float* pbuf = smem + 2048 + wave * 256;
  const int row0 = (blockIdx.x * 4 + wave) * 16;

  // Preload Q fragments (A-layout): 16 k-steps x float2 per lane.
  v2f qf[16];
  const float* qrow = Q + (size_t)(row0 + m) * DIM;
#pragma unroll
  for (int ks = 0; ks < 16; ++ks) qf[ks] = *(const v2f*)(qrow + 4 * ks + 2 * h);

  v8f oacc[4] = {};
  float Ms[8], Ls[8];
#pragma unroll
  for (int r = 0; r < 8; ++r) {
    Ms[r] = -INFINITY;
    Ls[r] = 0.0f;
  }

  for (int kb = 0; kb < n; kb += 16) {
    __syncthreads();  // staging buffers free to overwrite
    // A 16x64 tile is one contiguous 4 KB block: stage K and V tiles
    // cooperatively (4 x async b128 per thread).
    {
      const size_t gbase = (size_t)kb * DIM;
#pragma unroll
      for (int rr = 0; rr < 2; ++rr) {
        const int c = (tid + 128 * rr) * 4;
        stage16(K + gbase + c, kbuf + c);
        stage16(V + gbase + c, vbuf + c);
      }
    }
    stage_wait();
    __syncthreads();  // tiles visible to all waves

    // S tile = Q @ K^T : B frag element B[k][nk] = K[kb+nk][k] (ds reads)
    v8f s = {};
#pragma unroll
    for (int ks = 0; ks < 16; ++ks) {
      v2f kf = *(const v2f*)(kbuf + m * DIM + 4 * ks + 2 * h);
      s = wmma4(qf[ks], kf, s);
    }

    // Online softmax. Row (r + 8h) lives on the 16 lanes of half h in
    // accumulator element r -> reduce with xor shuffles 1,2,4,8.
    float corr[8];
#pragma unroll
    for (int r = 0; r < 8; ++r) {
      float mt = s[r];
      mt = fmaxf(mt, __shfl_xor(mt, 1, 32));
      mt = fmaxf(mt, __shfl_xor(mt, 2, 32));
      mt = fmaxf(mt, __shfl_xor(mt, 4, 32));
      mt = fmaxf(mt, __shfl_xor(mt, 8, 32));
      float newM = fmaxf(Ms[r], mt);
      corr[r] = __expf(Ms[r] - newM);
      Ms[r] = newM;
      float p = __expf(s[r] - newM);
      s[r] = p;
      float ts = p;
      ts += __shfl_xor(ts, 1, 32);
      ts += __shfl_xor(ts, 2, 32);
      ts += __shfl_xor(ts, 4, 32);
      ts += __shfl_xor(ts, 8, 32);
      Ls[r] = Ls[r] * corr[r] + ts;
    }
#pragma unroll
    for (int j = 0; j < 4; ++j) {
#pragma unroll
      for (int r = 0; r < 8; ++r) oacc[j][r] = oacc[j][r] * corr[r];
    }

    // Transpose P: C-layout (lane holds P[r+8h][m]) -> row-major LDS tile.
#pragma unroll
    for (int r = 0; r < 8; ++r) pbuf[(r + 8 * h) * 16 + m] = s[r];
    __syncthreads();  // P tiles ready (also orders intra-wave LDS use)

    // O += P @ V : A frag from pbuf, B frag from vbuf (all ds reads).
#pragma unroll
    for (int ks = 0; ks < 4; ++ks) {
      const int k0 = 4 * ks + 2 * h;
      v2f pf = *(const v2f*)(pbuf + m * 16 + k0);
#pragma unroll
      for (int j = 0; j < 4; ++j) {
        v2f vf;
        vf.x = vbuf[(k0 + 0) * DIM + 16 * j + m];
        vf.y = vbuf[(k0 + 1) * DIM + 16 * j + m];
        oacc[j] = wmma4(pf, vf, oacc[j]);
      }
    }
  }

#pragma unroll
  for (int r = 0; r < 8; ++r) {
    float inv = 1.0f / Ls[r];
#pragma unroll
    for (int j = 0; j < 4; ++j) {
      O[(size_t)(row0 + r + 8 * h) * DIM + 16 * j + m] = oacc[j][r] * inv;
    }
  }
}

// ---------------------------------------------------------------------------
// Gather + concat(h1,h2,ho) + tanh + p5-normalize. One 192-thread block/row.
// ---------------------------------------------------------------------------
__global__ __launch_bounds__(192) void finalize_kernel(
    const float* __restrict__ h1u, const float* __restrict__ h2u,
    const float* __restrict__ hou, const float* __restrict__ h1i,
    const float* __restrict__ h2i, const float* __restrict__ hoi,
    const int* __restrict__ users, const int* __restrict__ pos,
    const int* __restrict__ neg, const int* __restrict__ ousers,
    const int* __restrict__ opos, const int* __restrict__ oneg,
    float* __restrict__ out) {
  __shared__ float red[192];
  const int t = threadIdx.x;
  const int which = blockIdx.x / BATCH_SZ;
  const int b = blockIdx.x % BATCH_SZ;

  const float *h1, *h2, *ho;
  int idx, oidx;
  if (which == 0) {
    h1 = h1u; h2 = h2u; ho = hou;
    idx = users[b]; oidx = ousers[b];
  } else {
    h1 = h1i; h2 = h2i; ho = hoi;
    if (which == 1) { idx = pos[b]; oidx = opos[b]; }
    else            { idx = neg[b]; oidx = oneg[b]; }
  }

  const int seg = t >> 6;
  const int d = t & 63;
  float x;
  if (seg == 0)      x = h1[(size_t)idx * DIM + d];
  else if (seg == 1) x = h2[(size_t)idx * DIM + d];
  else               x = ho[(size_t)oidx * DIM + d];
  x = tanhf(x);

  float a = fabsf(x);
  red[t] = a * a * a * a * a;
  __syncthreads();
  if (t < 64) red[t] += red[t + 64] + red[t + 128];
  __syncthreads();
  if (t < 32) {
    float v = red[t] + red[t + 32];
    v += __shfl_xor(v, 16, 32);
    v += __shfl_xor(v, 8, 32);
    v += __shfl_xor(v, 4, 32);
    v += __shfl_xor(v, 2, 32);
    v += __shfl_xor(v, 1, 32);
    if (t == 0) red[0] = v;
  }
  __syncthreads();
  float norm = powf(red[0], 0.2f);
  out[(size_t)blockIdx.x * 192 + t] = x / fmaxf(norm, 1e-12f);
}

// ---------------------------------------------------------------------------
extern "C" void kernel_launch(void* const* d_in, const int* in_sizes, int n_in,
                              void* d_out, int out_size, void* d_ws,
                              size_t ws_size, hipStream_t stream) {
  const float* user_emb  = (const float*)d_in[0];
  const float* item_emb  = (const float*)d_in[1];
  const float* atten_w_u = (const float*)d_in[2];
  const float* atten_w_i = (const float*)d_in[3];
  const float* mean_w_u  = (const float*)d_in[4];
  const float* mean_w_i  = (const float*)d_in[5];
  const float* obs_w_u   = (const float*)d_in[6];
  const float* obs_w_i   = (const float*)d_in[7];
  const float* s_u_nj    = (const float*)d_in[8];
  const float* s_i_nj    = (const float*)d_in[9];
  const float* o_u_nj    = (const float*)d_in[10];
  const float* o_i_nj    = (const float*)d_in[11];
  const float* adj_val   = (const float*)d_in[12];
  const float* obs_val   = (const float*)d_in[13];
  const int* users    = (const int*)d_in[14];
  const int* pos      = (const int*)d_in[15];
  const int* neg      = (const int*)d_in[16];
  const int* ousers   = (const int*)d_in[17];
  const int* opos     = (const int*)d_in[18];
  const int* oneg     = (const int*)d_in[19];
  const int* adj_row  = (const int*)d_in[20];
  const int* adj_col  = (const int*)d_in[21];
  const int* obs_row  = (const int*)d_in[22];
  const int* obs_col  = (const int*)d_in[23];
  const int nedges  = in_sizes[12];
  const int nedgeso = in_sizes[13];

  const size_t F = (size_t)N_NODES * DIM;  // 524288 floats = 2 MB
  float* B    = (float*)d_ws;
  float* t_ej = B + 0 * F;   // projection scratch (e_j side)
  float* t_ek = B + 1 * F;   // projection scratch (e_k side / V)
  float* un   = B + 2 * F;   // spmm accumulators
  float* inb  = B + 3 * F;
  float* av   = B + 4 * F;   // attention output
  float* h1u  = B + 5 * F;
  float* h1i  = B + 6 * F;
  float* h2u  = B + 7 * F;
  float* h2i  = B + 8 * F;
  float* hou  = B + 9 * F;
  float* hoi  = B + 10 * F;

  const int ggrid = N_NODES / 128;  // gemm64: 8 waves x 16 rows
  const int zgrid = (int)(F / 256);
  const int sgrid = nedges * 32 / 256;
  const int sgrido = nedgeso * 32 / 256;

  // ---- user attention branch ----
  gemm64_kernel<0, 0><<<ggrid, 256, 0, stream>>>(user_emb, nullptr, atten_w_u, t_ej);
  gemm64_kernel<0, 0><<<ggrid, 256, 0, stream>>>(item_emb, nullptr, atten_w_u, t_ek);
  zero_kernel<<<zgrid, 256, 0, stream>>>(un, (int)F);
  spmm_kernel<<<sgrid, 256, 0, stream>>>(adj_row, adj_col, adj_val, t_ek, un, nedges);
  zero_kernel<<<zgrid, 256, 0, stream>>>(inb, (int)F);
  spmm_kernel<<<sgrid, 256, 0, stream>>>(adj_col, adj_row, adj_val, t_ej, inb, nedges);
  attn_kernel<<<N_NODES / 64, 128, 0, stream>>>(un, inb, t_ek, av, N_NODES);
  gemm64_kernel<0, 0><<<ggrid, 256, 0, stream>>>(av, nullptr, atten_w_u, h1u);

  // ---- item attention branch (roles swapped) ----
  gemm64_kernel<0, 0><<<ggrid, 256, 0, stream>>>(item_emb, nullptr, atten_w_i, t_ej);
  gemm64_kernel<0, 0><<<ggrid, 256, 0, stream>>>(user_emb, nullptr, atten_w_i, t_ek);
  zero_kernel<<<zgrid, 256, 0, stream>>>(un, (int)F);
  spmm_kernel<<<sgrid, 256, 0, stream>>>(adj_col, adj_row, adj_val, t_ek, un, nedges);
  zero_kernel<<<zgrid, 256, 0, stream>>>(inb, (int)F);
  spmm_kernel<<<sgrid, 256, 0, stream>>>(adj_row, adj_col, adj_val, t_ej, inb, nedges);
  attn_kernel<<<N_NODES / 64, 128, 0, stream>>>(un, inb, t_ek, av, N_NODES);
  gemm64_kernel<0, 0><<<ggrid, 256, 0, stream>>>(av, nullptr, atten_w_i, h1i);

  // ---- sampled-graph mean convs ----
  gemm64_kernel<0, 0><<<ggrid, 256, 0, stream>>>(item_emb, nullptr, mean_w_u, t_ej);
  zero_kernel<<<zgrid, 256, 0, stream>>>(un, (int)F);
  spmm_kernel<<<sgrid, 256, 0, stream>>>(adj_row, adj_col, adj_val, t_ej, un, nedges);
  gemm64_kernel<1, 0><<<ggrid, 256, 0, stream>>>(un, s_u_nj, mean_w_u, h2u);

  gemm64_kernel<0, 0><<<ggrid, 256, 0, stream>>>(user_emb, nullptr, mean_w_i, t_ej);
  zero_kernel<<<zgrid, 256, 0, stream>>>(un, (int)F);
  spmm_kernel<<<sgrid, 256, 0, stream>>>(adj_col, adj_row, adj_val, t_ej, un, nedges);
  gemm64_kernel<1, 0><<<ggrid, 256, 0, stream>>>(un, s_i_nj, mean_w_i, h2i);

  // ---- observed-graph mean convs (tanh epilogue) ----
  gemm64_kernel<0, 0><<<ggrid, 256, 0, stream>>>(item_emb, nullptr, obs_w_u, t_ej);
  zero_kernel<<<zgrid, 256, 0, stream>>>(un, (int)F);
  spmm_kernel<<<sgrido, 256, 0, stream>>>(obs_row, obs_col, obs_val, t_ej, un, nedgeso);
  gemm64_kernel<1, 1><<<ggrid, 256, 0, stream>>>(un, o_u_nj, obs_w_u, hou);

  gemm64_kernel<0, 0><<<ggrid, 256, 0, stream>>>(user_emb, nullptr, obs_w_i, t_ej);
  zero_kernel<<<zgrid, 256, 0, stream>>>(un, (int)F);
  spmm_kernel<<<sgrido, 256, 0, stream>>>(obs_col, obs_row, obs_val, t_ej, un, nedgeso);
  gemm64_kernel<1, 1><<<ggrid, 256, 0, stream>>>(un, o_i_nj, obs_w_i, hoi);

  // ---- gather / concat / tanh / p5-normalize ----
  finalize_kernel<<<3 * BATCH_SZ, 192, 0, stream>>>(
      h1u, h2u, hou, h1i, h2i, hoi, users, pos, neg, ousers, opos, oneg,
      (float*)d_out);
}